// DPMM_8830452761108
// MI455X (gfx1250) — compile-verified
//
#include <hip/hip_runtime.h>
#include <math.h>

typedef __attribute__((ext_vector_type(2))) float v2f;
typedef __attribute__((ext_vector_type(8))) float v8f;

// ---------------------------------------------------------------------------
// helpers
// ---------------------------------------------------------------------------
__device__ __forceinline__ float sel4(float4 v, int i) {
    float r = v.x;
    r = (i == 1) ? v.y : r;
    r = (i == 2) ? v.z : r;
    r = (i == 3) ? v.w : r;
    return r;
}

// feature k of row x: k<16 -> x[k/4]*x[k%4]; 16..19 -> x[k-16]; 20 -> 1; else 0
__device__ __forceinline__ float featval(float4 x, int k) {
    if (k < 16) return sel4(x, k >> 2) * sel4(x, k & 3);
    if (k < 20) return sel4(x, k - 16);
    return (k == 20) ? 1.0f : 0.0f;
}

__device__ double digamma_d(double x) {
    double r = 0.0;
    while (x < 6.0) { r -= 1.0 / x; x += 1.0; }
    double f = 1.0 / (x * x);
    r += log(x) - 0.5 / x
       - f * (1.0 / 12.0 - f * (1.0 / 120.0 - f * (1.0 / 252.0
         - f * (1.0 / 240.0 - f * (1.0 / 132.0)))));
    return r;
}

__device__ double invert4x4(const double* m, double* invOut) {
    double inv[16];
    inv[0]  =  m[5]*m[10]*m[15] - m[5]*m[11]*m[14] - m[9]*m[6]*m[15] + m[9]*m[7]*m[14] + m[13]*m[6]*m[11] - m[13]*m[7]*m[10];
    inv[4]  = -m[4]*m[10]*m[15] + m[4]*m[11]*m[14] + m[8]*m[6]*m[15] - m[8]*m[7]*m[14] - m[12]*m[6]*m[11] + m[12]*m[7]*m[10];
    inv[8]  =  m[4]*m[9]*m[15]  - m[4]*m[11]*m[13] - m[8]*m[5]*m[15] + m[8]*m[7]*m[13] + m[12]*m[5]*m[11] - m[12]*m[7]*m[9];
    inv[12] = -m[4]*m[9]*m[14]  + m[4]*m[10]*m[13] + m[8]*m[5]*m[14] - m[8]*m[6]*m[13] - m[12]*m[5]*m[10] + m[12]*m[6]*m[9];
    inv[1]  = -m[1]*m[10]*m[15] + m[1]*m[11]*m[14] + m[9]*m[2]*m[15] - m[9]*m[3]*m[14] - m[13]*m[2]*m[11] + m[13]*m[3]*m[10];
    inv[5]  =  m[0]*m[10]*m[15] - m[0]*m[11]*m[14] - m[8]*m[2]*m[15] + m[8]*m[3]*m[14] + m[12]*m[2]*m[11] - m[12]*m[3]*m[10];
    inv[9]  = -m[0]*m[9]*m[15]  + m[0]*m[11]*m[13] + m[8]*m[1]*m[15] - m[8]*m[3]*m[13] - m[12]*m[1]*m[11] + m[12]*m[3]*m[9];
    inv[13] =  m[0]*m[9]*m[14]  - m[0]*m[10]*m[13] - m[8]*m[1]*m[14] + m[8]*m[2]*m[13] + m[12]*m[1]*m[10] - m[12]*m[2]*m[9];
    inv[2]  =  m[1]*m[6]*m[15]  - m[1]*m[7]*m[14]  - m[5]*m[2]*m[15] + m[5]*m[3]*m[14] + m[13]*m[2]*m[7]  - m[13]*m[3]*m[6];
    inv[6]  = -m[0]*m[6]*m[15]  + m[0]*m[7]*m[14]  + m[4]*m[2]*m[15] - m[4]*m[3]*m[14] - m[12]*m[2]*m[7]  + m[12]*m[3]*m[6];
    inv[10] =  m[0]*m[5]*m[15]  - m[0]*m[7]*m[13]  - m[4]*m[1]*m[15] + m[4]*m[3]*m[13] + m[12]*m[1]*m[7]  - m[12]*m[3]*m[5];
    inv[14] = -m[0]*m[5]*m[14]  + m[0]*m[6]*m[13]  + m[4]*m[1]*m[14] - m[4]*m[2]*m[13] - m[12]*m[1]*m[6]  + m[12]*m[2]*m[5];
    inv[3]  = -m[1]*m[6]*m[11]  + m[1]*m[7]*m[10]  + m[5]*m[2]*m[11] - m[5]*m[3]*m[10] - m[9]*m[2]*m[7]   + m[9]*m[3]*m[6];
    inv[7]  =  m[0]*m[6]*m[11]  - m[0]*m[7]*m[10]  - m[4]*m[2]*m[11] + m[4]*m[3]*m[10] + m[8]*m[2]*m[7]   - m[8]*m[3]*m[6];
    inv[11] = -m[0]*m[5]*m[11]  + m[0]*m[7]*m[9]   + m[4]*m[1]*m[11] - m[4]*m[3]*m[9]  - m[8]*m[1]*m[7]   + m[8]*m[3]*m[5];
    inv[15] =  m[0]*m[5]*m[10]  - m[0]*m[6]*m[9]   - m[4]*m[1]*m[10] + m[4]*m[2]*m[9]  + m[8]*m[1]*m[6]   - m[8]*m[2]*m[5];
    double det = m[0]*inv[0] + m[1]*inv[4] + m[2]*inv[8] + m[3]*inv[12];
    double id = 1.0 / det;
    for (int i = 0; i < 16; ++i) invOut[i] = inv[i] * id;
    return det;
}

// ---------------------------------------------------------------------------
// K0: zero the 512-float stats accumulator
// ---------------------------------------------------------------------------
__global__ void dpmm_k0_zero(float* __restrict__ stats) {
    int t = blockIdx.x * blockDim.x + threadIdx.x;
    if (t < 512) stats[t] = 0.0f;
}

// ---------------------------------------------------------------------------
// K1: stats[16 x 32] = Phi^T @ [vec(x x^T)(16) | x(4),1,0...](16)  via WMMA f32 16x16x4
//     stats[t*32 + (4d+e)]  = S[t][d][e]
//     stats[t*32 + 16 + d]  = (X^T Phi)[d][t]
//     stats[t*32 + 20]      = Nk[t]
// ---------------------------------------------------------------------------
__global__ void dpmm_k1_stats(const float* __restrict__ data,
                              const float* __restrict__ Phi,
                              float* __restrict__ stats,
                              int nChunks)  // N/4
{
    __shared__ float sacc[512];
    const int tid = threadIdx.x;
    sacc[tid] = 0.0f;
    sacc[tid + 256] = 0.0f;
    __syncthreads();

    const int lane = tid & 31;
    const int half = lane >> 4;       // A/B K-pair select (K=0,1 vs K=2,3)
    const int l16  = lane & 15;       // A: M (=cluster), B: N (=feature col)
    const int gwave = blockIdx.x * 8 + (tid >> 5);
    const int nWaves = gridDim.x * 8;

    const int d = l16 >> 2, e = l16 & 3;
    const float4* dat4 = (const float4*)data;

    v8f acc0 = {};   // outer-product features
    v8f acc1 = {};   // [x, 1] features

    for (int c = gwave; c < nChunks; c += nWaves) {
        const long n0 = (long)c * 4 + 2 * half;  // this lane's data-row pair
        // A = Phi^T chunk: lane holds Phi[n0][l16] (K even) and Phi[n0+1][l16] (K odd)
        const float* pr = Phi + n0 * 16;
        v2f a; a.x = pr[l16]; a.y = pr[16 + l16];
        // B = feature rows of the same data rows (broadcast within 16-lane half)
        float4 x0 = dat4[n0];
        float4 x1 = dat4[n0 + 1];
        v2f b0; b0.x = sel4(x0, d) * sel4(x0, e);
                b0.y = sel4(x1, d) * sel4(x1, e);
        v2f b1;
        b1.x = (l16 < 4) ? sel4(x0, l16) : ((l16 == 4) ? 1.0f : 0.0f);
        b1.y = (l16 < 4) ? sel4(x1, l16) : ((l16 == 4) ? 1.0f : 0.0f);

        acc0 = __builtin_amdgcn_wmma_f32_16x16x4_f32(false, a, false, b0, (short)0, acc0, false, false);
        acc1 = __builtin_amdgcn_wmma_f32_16x16x4_f32(false, a, false, b1, (short)0, acc1, false, false);
    }

    // C layout: VGPR v, lane -> M = v + 8*(lane/16), N = lane%16
    #pragma unroll
    for (int v = 0; v < 8; ++v) {
        const int M = v + 8 * half;
        atomicAdd(&sacc[M * 32 + l16], acc0[v]);        // ds_add_f32
        atomicAdd(&sacc[M * 32 + 16 + l16], acc1[v]);
    }
    __syncthreads();
    atomicAdd(&stats[tid], sacc[tid]);                  // global_atomic_add_f32
    atomicAdd(&stats[tid + 256], sacc[tid + 256]);
}

// ---------------------------------------------------------------------------
// K2: tiny per-cluster math (T=16, D=4) in fp64 on one thread.
//     Emits W[24][16]: rows 0..15 = -0.5*Lam[t] (by 4d+e), 16..19 = eta2,
//     row 20 = per-cluster logit constant, rows 21..23 = 0.
// ---------------------------------------------------------------------------
__global__ void dpmm_k2_prep(const float* __restrict__ stats,
                             const float* __restrict__ priorMu,
                             const float* __restrict__ priorKappa,
                             const float* __restrict__ priorPsi,
                             const float* __restrict__ priorNu,
                             float* __restrict__ W)
{
    if (threadIdx.x != 0 || blockIdx.x != 0) return;

    double Nk[16];
    for (int t = 0; t < 16; ++t) Nk[t] = (double)stats[t * 32 + 20];

    const double k0  = (double)priorKappa[0];
    const double nu0 = (double)priorNu[0];
    double mu0[4];
    for (int i = 0; i < 4; ++i) mu0[i] = (double)priorMu[i];
    double prior11[16];
    for (int d = 0; d < 4; ++d)
        for (int e = 0; e < 4; ++e)
            prior11[d * 4 + e] = (double)priorPsi[d * 4 + e] + k0 * mu0[d] * mu0[e];

    // stick-breaking gammas
    double g1[16], g2[16];
    double tail = 0.0;
    for (int t = 15; t >= 0; --t) { g2[t] = 1e-3 + tail; tail += Nk[t]; g1[t] = 1.0 + Nk[t]; }
    double dg1[16], dg2[16];
    for (int t = 0; t < 16; ++t) {
        double ds = digamma_d(g1[t] + g2[t]);
        dg1[t] = digamma_d(g1[t]) - ds;
        dg2[t] = digamma_d(g2[t]) - ds;
    }

    const double LOG2 = 0.6931471805599453;
    double term2 = 0.0;
    for (int t = 0; t < 16; ++t) {
        const double kap = k0 + Nk[t];
        const double nu  = Nk[t] + nu0;
        double mu[4];
        for (int d = 0; d < 4; ++d)
            mu[d] = (k0 * mu0[d] + (double)stats[t * 32 + 16 + d]) / kap;
        double Psi[16];
        for (int d = 0; d < 4; ++d)
            for (int e = 0; e < 4; ++e)
                Psi[d * 4 + e] = prior11[d * 4 + e] + (double)stats[t * 32 + d * 4 + e]
                               - kap * mu[d] * mu[e];
        double Pinv[16];
        const double det = invert4x4(Psi, Pinv);
        double quad = 0.0, eta2[4];
        for (int d = 0; d < 4; ++d) {
            double s = 0.0;
            for (int e = 0; e < 4; ++e) s += Pinv[d * 4 + e] * mu[e];
            eta2[d] = nu * s;
            quad += mu[d] * s;
        }
        const double eta3 = -digamma_d(0.5 * nu) - 4.0 * LOG2 + log(det);
        const double eta4 = -0.5 * 4.0 / kap - 0.5 * nu * quad;
        const double cst  = dg1[t] + term2 - 0.5 * eta3 + eta4;
        term2 += dg2[t];

        for (int d = 0; d < 4; ++d)
            for (int e = 0; e < 4; ++e)
                W[(d * 4 + e) * 16 + t] = (float)(-0.5 * nu * Pinv[d * 4 + e]);
        for (int d = 0; d < 4; ++d) W[(16 + d) * 16 + t] = (float)eta2[d];
        W[20 * 16 + t] = (float)cst;
        for (int r = 21; r < 24; ++r) W[r * 16 + t] = 0.0f;
    }
}

// ---------------------------------------------------------------------------
// K3: logits[16 rows x 16 clusters] = Feat[16x24] @ W[24x16] via 6 WMMAs,
//     then a 16-lane softmax per row; one 16-row tile per wave.
// ---------------------------------------------------------------------------
__global__ void dpmm_k3_estep(const float* __restrict__ data,
                              const float* __restrict__ W,
                              float* __restrict__ out,
                              int nTiles)  // N/16
{
    const int tid  = threadIdx.x;
    const int lane = tid & 31;
    const int half = lane >> 4;
    const int l16  = lane & 15;
    const int tile = blockIdx.x * 8 + (tid >> 5);
    if (tile >= nTiles) return;   // wave-uniform

    // B (weights), chunk c covers features k=4c..4c+3
    v2f b[6];
    #pragma unroll
    for (int c = 0; c < 6; ++c) {
        const int k = 4 * c + 2 * half;
        b[c].x = W[k * 16 + l16];
        b[c].y = W[(k + 1) * 16 + l16];
    }

    const long n0 = (long)tile * 16;
    const float4 x = ((const float4*)data)[n0 + l16];  // this lane's row (both halves same row)

    v8f acc = {};
    #pragma unroll
    for (int c = 0; c < 6; ++c) {
        const int k = 4 * c + 2 * half;
        v2f a; a.x = featval(x, k); a.y = featval(x, k + 1);
        acc = __builtin_amdgcn_wmma_f32_16x16x4_f32(false, a, false, b[c], (short)0, acc, false, false);
    }

    // softmax over clusters: clusters are spread across the 16 lanes of each half
    #pragma unroll
    for (int v = 0; v < 8; ++v) {
        float val = acc[v];
        float m = val;
        #pragma unroll
        for (int off = 1; off < 16; off <<= 1)
            m = fmaxf(m, __shfl_xor(m, off, 32));
        float ex = __expf(val - m);
        float s = ex;
        #pragma unroll
        for (int off = 1; off < 16; off <<= 1)
            s += __shfl_xor(s, off, 32);
        out[(n0 + v + 8 * half) * 16 + l16] = ex / s;   // 64B-coalesced per half
    }
}

// ---------------------------------------------------------------------------
extern "C" void kernel_launch(void* const* d_in, const int* in_sizes, int n_in,
                              void* d_out, int out_size, void* d_ws, size_t ws_size,
                              hipStream_t stream) {
    const float* data       = (const float*)d_in[0];   // [N,4]
    const float* Phi        = (const float*)d_in[1];   // [N,16]
    const float* priorMu    = (const float*)d_in[2];   // [4]
    const float* priorKappa = (const float*)d_in[3];   // [1]
    const float* priorPsi   = (const float*)d_in[4];   // [16]
    const float* priorNu    = (const float*)d_in[5];   // [1]
    float* out = (float*)d_out;

    const int N = in_sizes[0] / 4;     // D = 4
    float* stats = (float*)d_ws;       // 512 floats
    float* W     = stats + 512;        // 384 floats

    dpmm_k0_zero<<<2, 256, 0, stream>>>(stats);
    dpmm_k1_stats<<<256, 256, 0, stream>>>(data, Phi, stats, N / 4);
    dpmm_k2_prep<<<1, 32, 0, stream>>>(stats, priorMu, priorKappa, priorPsi, priorNu, W);
    const int nTiles = N / 16;
    dpmm_k3_estep<<<(nTiles + 7) / 8, 256, 0, stream>>>(data, W, out, nTiles);
}